// GAT_81767587381366
// MI455X (gfx1250) — compile-verified
//
#include <hip/hip_runtime.h>
#include <cstdint>

// GAT on MI455X (gfx1250, wave32).
// - GEMMs use V_WMMA_F32_16X16X4_F32 (native fp32 WMMA -> no precision loss).
// - W operand is DMA'd into LDS by the Tensor Data Mover (tensor_load_to_lds,
//   TENSORcnt), A tile staged cooperatively; fragments are fed from LDS.
// - Edge softmax/aggregation uses fp32 atomics; whole working set (~85MB)
//   is resident in the 192MB L2, so atomics are L2-local RMWs.

#define FDIM 128
#define HH   8
#define CC   16
#define FOUT 16
#define NEG_SLOPE 0.2f

typedef float v2f __attribute__((ext_vector_type(2)));
typedef float v8f __attribute__((ext_vector_type(8)));
typedef unsigned int u32x4 __attribute__((ext_vector_type(4)));
typedef int i32x4 __attribute__((ext_vector_type(4)));
typedef int i32x8 __attribute__((ext_vector_type(8)));

#if __has_builtin(__builtin_amdgcn_tensor_load_to_lds) && \
    __has_builtin(__builtin_amdgcn_s_wait_tensorcnt)
#define USE_TDM 1
#else
#define USE_TDM 0
#endif

// ---------- order-preserving float<->int encoding for atomicMax ----------
__device__ __forceinline__ int enc_f32(float x) {
  int iv = __float_as_int(x);
  return (iv >= 0) ? iv : (iv ^ 0x7fffffff);
}
__device__ __forceinline__ float dec_f32(int e) {
  return __int_as_float((e >= 0) ? e : (e ^ 0x7fffffff));
}

// ---------- fills ----------
__global__ void fill_f32(float* __restrict__ p, float v, int n) {
  int i = blockIdx.x * blockDim.x + threadIdx.x;
  if (i < n) p[i] = v;
}
__global__ void fill_i32(int* __restrict__ p, int v, int n) {
  int i = blockIdx.x * blockDim.x + threadIdx.x;
  if (i < n) p[i] = v;
}

// ---------- fp32 WMMA GEMM: C[M,Ncols] = A[M,K] @ B[K,Ncols] (+bias, act) ----------
// One wave computes one 16x16 tile of C. blockDim.x = (Ncols/16)*32, wave w -> col tile w.
// grid.x = M/16 (M = 50000 = 3125*16, exact).
// LDS: W staged (TDM if available) with row stride Ncols+8 (bank-conflict pad,
//      half-waves read rows k and k+2 -> disjoint bank ranges);
//      A tile (16 x K) staged with row stride K+4 (lanes -> banks 4*i).
// A-fragment (16x4 f32): lane L holds row M=L%16, K = 2*(L/16) + {vgpr0,vgpr1}.
// B-fragment (4x16 f32): lane L holds col N=L%16, K = 2*(L/16) + {vgpr0,vgpr1}.
// C/D (16x16 f32): lane L, vgpr g -> row M = g + 8*(L/16), col N = L%16.
// act: 0 = none, 2 = sigmoid (final Wout projection).
__global__ void gemm_wmma_f32(const float* __restrict__ A, const float* __restrict__ B,
                              const float* __restrict__ bias, float* __restrict__ C,
                              int K, int Ncols, int act) {
  extern __shared__ float smem[];
  const int wls = Ncols + 8;                 // padded W row stride (dwords)
  const int als = K + 4;                     // padded A row stride (dwords)
  float* Wl = smem;                          // K * wls
  float* Al = smem + (size_t)K * wls;        // 16 * als

  const int wave = threadIdx.x >> 5;
  const int lane = threadIdx.x & 31;
  const int row0 = blockIdx.x * 16;
  const int col0 = wave * 16;
  const int mA   = lane & 15;
  const int hi   = lane >> 4;                // 0 or 1
  const int kHi  = hi * 2;
  const int nB   = lane & 15;

#if USE_TDM
  if (wave == 0) {
    // ---- Tensor DMA descriptor (D#): 2D tile, whole W (K rows x Ncols), 4B elems,
    // ---- LDS padding: +8 dwords after every Ncols dwords -> LDS row stride wls.
    const unsigned ldsW = (unsigned)(size_t)(void*)Wl;   // low 32 bits = LDS byte addr
    const unsigned long long ga = (unsigned long long)(size_t)B;
    const int piv = 31 - __clz((unsigned)Ncols) - 1;     // pad_interval: 2^(v+1) dwords = Ncols

    u32x4 g0;
    g0.x = 1u;                                           // count=1 (valid), user mode
    g0.y = ldsW;                                         // lds_addr [63:32]
    g0.z = (unsigned)ga;                                 // global_addr [95:64]
    g0.w = (unsigned)((ga >> 32) & 0x01ffffffull)        // global_addr [120:96]
           | (2u << 30);                                 // type=2 ("image")
    i32x8 g1;
    g1[0] = (2 << 16)                                    // data_size = 4B
          | (1 << 20)                                    // pad_enable
          | (piv << 22)                                  // pad_interval
          | (7 << 25);                                   // pad_amount = 8 dwords
    g1[1] = (Ncols & 0xffff) << 16;                      // tensor_dim0[15:0]  (bits 63:48)
    g1[2] = ((Ncols >> 16) & 0xffff)                     // tensor_dim0[31:16]
          | ((K & 0xffff) << 16);                        // tensor_dim1[15:0]
    g1[3] = ((K >> 16) & 0xffff)                         // tensor_dim1[31:16]
          | ((Ncols & 0xffff) << 16);                    // tile_dim0
    g1[4] = (K & 0xffff);                                // tile_dim1 ; tile_dim2=0
    g1[5] = Ncols;                                       // tensor_dim0_stride[31:0]
    g1[6] = 0;
    g1[7] = 0;
    i32x4 z4 = {};
#if __clang_major__ >= 23
    i32x8 z8 = {};
    __builtin_amdgcn_tensor_load_to_lds(g0, g1, z4, z4, z8, 0);
#else
    __builtin_amdgcn_tensor_load_to_lds(g0, g1, z4, z4, 0);
#endif
  }
#else
  for (int i = threadIdx.x; i < K * Ncols; i += blockDim.x) {
    int r = i / Ncols, c = i % Ncols;
    Wl[r * wls + c] = B[i];
  }
#endif

  // cooperative stage of the 16 x K A tile (contiguous rows of A)
  {
    const float* Ablk = A + (size_t)row0 * K;
    for (int i = threadIdx.x; i < 16 * K; i += blockDim.x) {
      int r = i / K, c = i % K;
      Al[r * als + c] = Ablk[i];
    }
  }

#if USE_TDM
  if (wave == 0) __builtin_amdgcn_s_wait_tensorcnt(0);
#endif
  __syncthreads();

  const float* Arow = Al + (size_t)mA * als;
  v8f acc = {};
  for (int k = 0; k < K; k += 4) {
    v2f a, b;
    a.x = Arow[k + kHi];
    a.y = Arow[k + kHi + 1];
    b.x = Wl[(k + kHi)     * wls + col0 + nB];
    b.y = Wl[(k + kHi + 1) * wls + col0 + nB];
    acc = __builtin_amdgcn_wmma_f32_16x16x4_f32(
        /*neg_a=*/false, a, /*neg_b=*/false, b,
        /*c_mod=*/(short)0, acc, /*reuse_a=*/false, /*reuse_b=*/false);
  }

  const int col = col0 + nB;
  #pragma unroll
  for (int g = 0; g < 8; ++g) {
    int m = g + hi * 8;
    float v = acc[g];
    if (bias) v += bias[col];
    if (act == 2) v = 1.0f / (1.0f + __expf(-v));
    C[(size_t)(row0 + m) * Ncols + col] = v;
  }
}

// ---------- per-node attention logits: al_src/al_dst [n,H] ----------
__global__ void attn_logits(const float* __restrict__ Hf,
                            const float* __restrict__ a_src, const float* __restrict__ a_dst,
                            float* __restrict__ alS, float* __restrict__ alD, int n) {
  int i = blockIdx.x * blockDim.x + threadIdx.x;
  if (i >= n * HH) return;
  int node = i / HH, h = i % HH;
  const float* hp = Hf + (size_t)node * FDIM + h * CC;
  float s = 0.f, d = 0.f;
  #pragma unroll
  for (int c = 0; c < CC; ++c) {
    float v = hp[c];
    s += v * a_src[h * CC + c];
    d += v * a_dst[h * CC + c];
  }
  alS[i] = s;
  alD[i] = d;
}

// ---------- edge pass 1: leaky-relu score + segment max (atomic) ----------
__global__ void edge_max(const float* __restrict__ alS, const float* __restrict__ alD,
                         const int* __restrict__ srcI, const int* __restrict__ dstI,
                         float* __restrict__ ebuf, int* __restrict__ maxb, int E, int n) {
  long long i = (long long)blockIdx.x * blockDim.x + threadIdx.x;
  long long MT = (long long)E + n;
  if (i >= MT * HH) return;
  int e = (int)(i / HH), h = (int)(i % HH);
  int s, d;
  if (e < E) { s = srcI[e]; d = dstI[e]; } else { s = d = e - E; }  // self-loops
  float v = alS[(size_t)s * HH + h] + alD[(size_t)d * HH + h];
  v = (v > 0.f) ? v : NEG_SLOPE * v;
  ebuf[i] = v;
  atomicMax(&maxb[(size_t)d * HH + h], enc_f32(v));
}

// ---------- edge pass 2: exp(e - max) + segment sum (atomic) ----------
__global__ void edge_exp(float* __restrict__ ebuf, const int* __restrict__ dstI,
                         const int* __restrict__ maxb, float* __restrict__ sumb, int E, int n) {
  long long i = (long long)blockIdx.x * blockDim.x + threadIdx.x;
  long long MT = (long long)E + n;
  if (i >= MT * HH) return;
  int e = (int)(i / HH), h = (int)(i % HH);
  int d = (e < E) ? dstI[e] : (e - E);
  float m  = dec_f32(maxb[(size_t)d * HH + h]);
  float ex = __expf(ebuf[i] - m);
  ebuf[i] = ex;
  atomicAdd(&sumb[(size_t)d * HH + h], ex);
}

// ---------- edge pass 3: agg[dst] += h[src] * alpha ----------
__global__ void edge_aggr(const float* __restrict__ Hf, const float* __restrict__ ebuf,
                          const float* __restrict__ sumb,
                          const int* __restrict__ srcI, const int* __restrict__ dstI,
                          float* __restrict__ agg, int E, int n) {
  long long i = (long long)blockIdx.x * blockDim.x + threadIdx.x;
  long long MT = (long long)E + n;
  if (i >= MT * FDIM) return;
  int e = (int)(i / FDIM), f = (int)(i % FDIM);
  int h = f / CC;
  int s, d;
  if (e < E) { s = srcI[e]; d = dstI[e]; } else { s = d = e - E; }
  float alpha = ebuf[(size_t)e * HH + h] / (sumb[(size_t)d * HH + h] + 1e-16f);
  atomicAdd(&agg[(size_t)d * FDIM + f], Hf[(size_t)s * FDIM + f] * alpha);
}

// ---------- bias + ELU (in place) ----------
__global__ void bias_elu(float* __restrict__ agg, const float* __restrict__ b, int total) {
  int i = blockIdx.x * blockDim.x + threadIdx.x;
  if (i >= total) return;
  float v = agg[i] + b[i % FDIM];
  agg[i] = (v > 0.f) ? v : (__expf(v) - 1.0f);
}

extern "C" void kernel_launch(void* const* d_in, const int* in_sizes, int n_in,
                              void* d_out, int out_size, void* d_ws, size_t ws_size,
                              hipStream_t stream) {
  const float* x     = (const float*)d_in[0];
  const int*   ei    = (const int*)  d_in[1];
  const float* W1    = (const float*)d_in[2];
  const float* as1   = (const float*)d_in[3];
  const float* ad1   = (const float*)d_in[4];
  const float* b1    = (const float*)d_in[5];
  const float* W2    = (const float*)d_in[6];
  const float* as2   = (const float*)d_in[7];
  const float* ad2   = (const float*)d_in[8];
  const float* b2    = (const float*)d_in[9];
  const float* Wout  = (const float*)d_in[10];
  const float* bout  = (const float*)d_in[11];

  const int n = in_sizes[0] / FDIM;      // 50000 (multiple of 16)
  const int E = in_sizes[1] / 2;         // 800000
  const int* srcI = ei;
  const int* dstI = ei + E;
  const long long MT = (long long)E + n; // with self-loops

  // carve workspace
  char* ws = (char*)d_ws;
  float* H1   = (float*)ws; ws += (size_t)n * FDIM * sizeof(float);
  float* AGG  = (float*)ws; ws += (size_t)n * FDIM * sizeof(float);
  float* ALS  = (float*)ws; ws += (size_t)n * HH * sizeof(float);
  float* ALD  = (float*)ws; ws += (size_t)n * HH * sizeof(float);
  int*   MAXB = (int*)  ws; ws += (size_t)n * HH * sizeof(int);
  float* SUMB = (float*)ws; ws += (size_t)n * HH * sizeof(float);
  float* EBUF = (float*)ws; ws += (size_t)MT * HH * sizeof(float);
  (void)ws_size; (void)n_in; (void)out_size;

  auto nb = [](long long t) { return (unsigned)((t + 255) / 256); };
  // dynamic LDS: W (K rows * (Ncols+8)) + A (16 * (K+4)), fp32
  auto lds_bytes = [](int K, int Ncols) {
    return (size_t)(K * (Ncols + 8) + 16 * (K + 4)) * sizeof(float);
  };

  auto gat_layer = [&](const float* xin, const float* W, const float* as,
                       const float* ad, const float* b) {
    // h = xin @ W  (N x 128 @ 128 x 128), fp32 WMMA, W via TDM->LDS
    gemm_wmma_f32<<<n / 16, (FDIM / 16) * 32, lds_bytes(FDIM, FDIM), stream>>>(
        xin, W, nullptr, H1, FDIM, FDIM, 0);
    attn_logits<<<nb((long long)n * HH), 256, 0, stream>>>(H1, as, ad, ALS, ALD, n);
    fill_i32<<<nb((long long)n * HH), 256, 0, stream>>>(MAXB, (int)0x80000000, n * HH);
    fill_f32<<<nb((long long)n * HH), 256, 0, stream>>>(SUMB, 0.f, n * HH);
    fill_f32<<<nb((long long)n * FDIM), 256, 0, stream>>>(AGG, 0.f, n * FDIM);
    edge_max <<<nb(MT * HH),   256, 0, stream>>>(ALS, ALD, srcI, dstI, EBUF, MAXB, E, n);
    edge_exp <<<nb(MT * HH),   256, 0, stream>>>(EBUF, dstI, MAXB, SUMB, E, n);
    edge_aggr<<<nb(MT * FDIM), 256, 0, stream>>>(H1, EBUF, SUMB, srcI, dstI, AGG, E, n);
    bias_elu <<<nb((long long)n * FDIM), 256, 0, stream>>>(AGG, b, n * FDIM);
  };

  gat_layer(x,   W1, as1, ad1, b1);   // AGG now holds ELU(gat1)
  gat_layer(AGG, W2, as2, ad2, b2);   // gemm consumes AGG first, then AGG re-zeroed & reused

  // final: sigmoid(AGG @ Wout + bout) -> d_out  (N x 16), sigmoid fused in epilogue
  gemm_wmma_f32<<<n / 16, (FOUT / 16) * 32, lds_bytes(FDIM, FOUT), stream>>>(
      AGG, Wout, bout, (float*)d_out, FDIM, FOUT, 2);
}